// SplatfactoModel_58884001628698
// MI455X (gfx1250) — compile-verified
//
#include <hip/hip_runtime.h>
#include <hip/hip_bf16.h>
#include <math.h>

typedef __attribute__((ext_vector_type(2))) float v2f;
typedef __attribute__((ext_vector_type(8))) float v8f;

#define TPB 256
#define LN255 5.545177444479562f

// ---------------------------------------------------------------------------
// Kernel 1: per-Gaussian preprocessing.
// Produces: zbuf[N] (depth key), coef[N][8] (sigma-GEMM coefficients),
//           colbuf[N][4] (rgb + effective opacity).
// ---------------------------------------------------------------------------
__global__ __launch_bounds__(TPB) void gs_preprocess(
    const float* __restrict__ means, const float* __restrict__ scales,
    const float* __restrict__ quats, const float* __restrict__ fdc,
    const float* __restrict__ frest, const float* __restrict__ opac,
    const float* __restrict__ viewmat, const float* __restrict__ Km,
    float* __restrict__ zbuf, float* __restrict__ coef,
    float* __restrict__ colbuf, int N)
{
    int i = blockIdx.x * blockDim.x + threadIdx.x;
    if (i >= N) return;

    float R[3][3], tc[3];
#pragma unroll
    for (int r = 0; r < 3; ++r) {
        R[r][0] = viewmat[r*4+0]; R[r][1] = viewmat[r*4+1];
        R[r][2] = viewmat[r*4+2]; tc[r]   = viewmat[r*4+3];
    }
    float fx = Km[0], cx = Km[2], fy = Km[4], cy = Km[5];

    float mean[3] = { means[3*i], means[3*i+1], means[3*i+2] };
    float pc[3];
#pragma unroll
    for (int r = 0; r < 3; ++r)
        pc[r] = R[r][0]*mean[0] + R[r][1]*mean[1] + R[r][2]*mean[2] + tc[r];
    float z = pc[2];
    bool valid = z > 0.01f;
    float zc = fmaxf(z, 0.01f);

    // quaternion -> rotation
    float qw = quats[4*i], qx = quats[4*i+1], qy = quats[4*i+2], qz = quats[4*i+3];
    float qn = rsqrtf(qw*qw + qx*qx + qy*qy + qz*qz);
    qw *= qn; qx *= qn; qy *= qn; qz *= qn;
    float Rg[3][3] = {
        {1.f-2.f*(qy*qy+qz*qz), 2.f*(qx*qy-qw*qz),    2.f*(qx*qz+qw*qy)},
        {2.f*(qx*qy+qw*qz),     1.f-2.f*(qx*qx+qz*qz), 2.f*(qy*qz-qw*qx)},
        {2.f*(qx*qz-qw*qy),     2.f*(qy*qz+qw*qx),    1.f-2.f*(qx*qx+qy*qy)}};
    float S[3] = { __expf(scales[3*i]), __expf(scales[3*i+1]), __expf(scales[3*i+2]) };

    float Mm[3][3];
#pragma unroll
    for (int r = 0; r < 3; ++r)
#pragma unroll
        for (int c = 0; c < 3; ++c) Mm[r][c] = Rg[r][c] * S[c];

    float c3[3][3];
#pragma unroll
    for (int r = 0; r < 3; ++r)
#pragma unroll
        for (int c = 0; c < 3; ++c)
            c3[r][c] = Mm[r][0]*Mm[c][0] + Mm[r][1]*Mm[c][1] + Mm[r][2]*Mm[c][2];

    float Tm[3][3];
#pragma unroll
    for (int r = 0; r < 3; ++r)
#pragma unroll
        for (int c = 0; c < 3; ++c)
            Tm[r][c] = R[r][0]*c3[0][c] + R[r][1]*c3[1][c] + R[r][2]*c3[2][c];
    float cc[3][3];
#pragma unroll
    for (int r = 0; r < 3; ++r)
#pragma unroll
        for (int c = 0; c < 3; ++c)
            cc[r][c] = Tm[r][0]*R[c][0] + Tm[r][1]*R[c][1] + Tm[r][2]*R[c][2];

    float izc = 1.f / zc;
    float j00 = fx * izc, j02 = -fx * pc[0] * izc * izc;
    float j11 = fy * izc, j12 = -fy * pc[1] * izc * izc;
    float v0[3], v1[3];
#pragma unroll
    for (int c = 0; c < 3; ++c) {
        v0[c] = j00*cc[0][c] + j02*cc[2][c];
        v1[c] = j11*cc[1][c] + j12*cc[2][c];
    }
    float a  = v0[0]*j00 + v0[2]*j02 + 0.3f;
    float b  = v0[1]*j11 + v0[2]*j12;
    float cf = v1[1]*j11 + v1[2]*j12 + 0.3f;
    float det = a*cf - b*b;
    float det_safe = (det > 0.f) ? det : 1.f;
    float inv = 1.f / det_safe;
    float cA = cf*inv, cB = -b*inv, cC = a*inv;
    float u = fx*pc[0]*izc + cx;
    float v = fy*pc[1]*izc + cy;

    // view direction for SH
    float cp[3];
#pragma unroll
    for (int r = 0; r < 3; ++r)
        cp[r] = -(R[0][r]*tc[0] + R[1][r]*tc[1] + R[2][r]*tc[2]);
    float dv[3] = { mean[0]-cp[0], mean[1]-cp[1], mean[2]-cp[2] };
    float dn = rsqrtf(dv[0]*dv[0] + dv[1]*dv[1] + dv[2]*dv[2]);
    float x = dv[0]*dn, y = dv[1]*dn, zd = dv[2]*dn;
    float xx = x*x, yy = y*y, zz = zd*zd;
    float bsh[16] = {
        0.28209479177387814f,
        -0.4886025119029199f*y,
        0.4886025119029199f*zd,
        -0.4886025119029199f*x,
        1.0925484305920792f*x*y,
        -1.0925484305920792f*y*zd,
        0.31539156525252005f*(2.f*zz-xx-yy),
        -1.0925484305920792f*x*zd,
        0.5462742152960396f*(xx-yy),
        -0.5900435899266435f*y*(3.f*xx-yy),
        2.890611442640554f*x*y*zd,
        -0.4570457994644658f*y*(4.f*zz-xx-yy),
        0.37317633259011546f*zd*(2.f*zz-3.f*xx-3.f*yy),
        -0.4570457994644658f*x*(4.f*zz-xx-yy),
        1.445305721320277f*zd*(xx-yy),
        -0.5900435899266435f*x*(xx-3.f*yy) };

    float rgb[3];
#pragma unroll
    for (int c = 0; c < 3; ++c) {
        float val = bsh[0] * fdc[3*i+c];
#pragma unroll
        for (int k = 1; k < 16; ++k)
            val += bsh[k] * frest[45*i + (k-1)*3 + c];
        rgb[c] = fmaxf(val + 0.5f, 0.f);
    }
    float so  = 1.f / (1.f + __expf(-opac[i]));
    float eff = (valid && det > 0.f) ? so : 0.f;

    zbuf[i] = zc;
    coef[8*i+0] = 0.5f*cA;
    coef[8*i+1] = 0.5f*cC;
    coef[8*i+2] = cB;
    coef[8*i+3] = -(cA*u + cB*v);
    coef[8*i+4] = -(cC*v + cB*u);
    coef[8*i+5] = 0.5f*cA*u*u + 0.5f*cC*v*v + cB*u*v;
    coef[8*i+6] = 0.f;
    coef[8*i+7] = 0.f;
    colbuf[4*i+0] = rgb[0]; colbuf[4*i+1] = rgb[1];
    colbuf[4*i+2] = rgb[2]; colbuf[4*i+3] = eff;
}

// ---------------------------------------------------------------------------
// Kernel 2: stable rank sort by depth (N <= 1024, one block) + gather.
// ---------------------------------------------------------------------------
__global__ __launch_bounds__(1024) void gs_sort_gather(
    const float* __restrict__ zbuf, const float* __restrict__ coef,
    const float* __restrict__ colbuf, float* __restrict__ scoef,
    float* __restrict__ scol, int N)
{
    __shared__ float zs[1024];
    int tid = threadIdx.x;
    zs[tid] = (tid < N) ? zbuf[tid] : 3.4e38f;
    __syncthreads();
    if (tid < N) {
        float zi = zs[tid];
        int rank = 0;
        for (int j = 0; j < N; ++j) {
            float zj = zs[j];
            rank += ((zj < zi) || (zj == zi && j < tid)) ? 1 : 0;
        }
        const float4* c4 = (const float4*)coef;
        float4*       s4 = (float4*)scoef;
        s4[rank*2+0] = c4[tid*2+0];
        s4[rank*2+1] = c4[tid*2+1];
        ((float4*)scol)[rank] = ((const float4*)colbuf)[tid];
    }
}

// ---------------------------------------------------------------------------
// Kernel 3: tiled rasterizer. 16x16-pixel tile per block (256 threads,
// 8 waves). Each wave owns 2 rows of 16 pixels. Sigma for each
// (16 pixel x 16 gaussian) tile is a rank-8 f32 GEMM via two chained
// V_WMMA_F32_16X16X4_F32. A wave-level __any vote on min(sigma) culls
// chunks whose gaussians cannot reach alpha >= 1/255 anywhere in the
// wave's footprint (sigma > ln 255). Surviving tiles are transposed
// through padded (bank-conflict-free) LDS so each lane owns one pixel
// for the sequential front-to-back compositing.
// ---------------------------------------------------------------------------
__global__ __launch_bounds__(TPB) void gs_raster(
    const float* __restrict__ scoef, const float* __restrict__ scol,
    const float* __restrict__ bg, float* __restrict__ out,
    int N, int W, int H)
{
    __shared__ float  sC[16][8];            // staged coefs for current chunk
    __shared__ float4 sColLds[16];          // staged rgb+opac
    __shared__ float  sigT[8][2][17][20];   // padded transpose [g][n][m]

    int tid  = threadIdx.x;
    int wave = tid >> 5;
    int lane = tid & 31;
    int half = lane >> 4;   // 0: lanes 0-15 (K0..1), 1: lanes 16-31 (K2..3)
    int m16  = lane & 15;

    int col     = blockIdx.x * 16 + m16;
    int rowBase = blockIdx.y * 16 + wave * 2;

    // Pixel-feature A matrices (constant over chunks), one per pixel group.
    float gx = (float)col + 0.5f;
    v2f A0[2], A1[2];
#pragma unroll
    for (int g = 0; g < 2; ++g) {
        float gy = (float)(rowBase + g) + 0.5f;
        float f[8] = { gx*gx, gy*gy, gx*gy, gx, gy, 1.f, 0.f, 0.f };
        int k0 = half * 2;
        A0[g][0] = f[k0];     A0[g][1] = f[k0+1];
        A1[g][0] = f[k0+4];   A1[g][1] = f[k0+5];
    }

    float Tacc = 1.f, cr = 0.f, cg = 0.f, cb = 0.f;
    int g_own = half, m_own = m16;

    int nChunks = (N + 15) >> 4;
    for (int c = 0; c < nChunks; ++c) {
        __syncthreads();   // previous chunk fully composited
        int base = c << 4;
        if (tid < 32) {
            // 16 gaussians x 8 coefs = 32 float4s
            int gi = base + (tid >> 1);
            float4 cv = make_float4(0.f, 0.f, 0.f, 0.f);
            if (gi < N) cv = ((const float4*)scoef)[base*2 + tid];
            ((float4*)sC)[tid] = cv;
        } else if (tid < 48) {
            int n = tid - 32, gi = base + n;
            float4 cv = make_float4(0.f, 0.f, 0.f, 0.f);
            if (gi < N) cv = ((const float4*)scol)[gi];
            sColLds[n] = cv;
        } else if (tid == 48 && base + 16 < N) {
            // prefetch next chunk into cache (global_prefetch_b8)
            __builtin_prefetch(scoef + (base + 16) * 8, 0, 0);
        }
        __syncthreads();

        // Gaussian-coefficient B matrices for this chunk.
        int k0 = half * 2;
        v2f B0, B1;
        B0[0] = sC[m16][k0];     B0[1] = sC[m16][k0+1];
        B1[0] = sC[m16][k0+4];   B1[1] = sC[m16][k0+5];

        // Sigma GEMM for both pixel groups.
        v8f d[2];
#pragma unroll
        for (int g = 0; g < 2; ++g) {
            v8f acc = {};
            acc = __builtin_amdgcn_wmma_f32_16x16x4_f32(
                      false, A0[g], false, B0, (short)0, acc, false, false);
            acc = __builtin_amdgcn_wmma_f32_16x16x4_f32(
                      false, A1[g], false, B1, (short)0, acc, false, false);
            d[g] = acc;
        }

        // Wave-level cull: if every sigma in this wave's 16x2-pixel
        // footprint exceeds ln(255), no gaussian in the chunk can reach
        // alpha >= 1/255 here -> skip transpose + compositing entirely.
        float dmin = d[0][0];
#pragma unroll
        for (int j = 1; j < 8; ++j) dmin = fminf(dmin, d[0][j]);
#pragma unroll
        for (int j = 0; j < 8; ++j) dmin = fminf(dmin, d[1][j]);
        if (__any(dmin < LN255)) {
            // lane holds D[m = r + half*8][n = m16]; store transposed rows.
#pragma unroll
            for (int g = 0; g < 2; ++g) {
                float* dst = &sigT[wave][g][m16][half * 8];
                ((float4*)dst)[0] = make_float4(d[g][0], d[g][1], d[g][2], d[g][3]);
                ((float4*)dst)[1] = make_float4(d[g][4], d[g][5], d[g][6], d[g][7]);
            }
            // sigT is wave-private and LDS ops of a wave are in-order:
            // no block barrier needed before reading it back.
#pragma unroll
            for (int n = 0; n < 16; ++n) {
                float sig  = sigT[wave][g_own][n][m_own];
                float4 cv  = sColLds[n];
                float em   = __expf(-fmaxf(sig, 0.f));
                float alpha = fminf(cv.w * em, 0.999f);
                alpha = (sig >= 0.f) ? alpha : 0.f;               // v_cndmask
                alpha = (alpha >= (1.f / 255.f)) ? alpha : 0.f;   // v_cndmask
                float wgt = Tacc * alpha;
                cr += wgt * cv.x; cg += wgt * cv.y; cb += wgt * cv.z;
                Tacc *= (1.f - alpha);
            }
        }
    }

    int row = rowBase + g_own;
    if (row < H && col < W) {
        int o = (row * W + col) * 3;
        out[o+0] = cr + Tacc * bg[0];
        out[o+1] = cg + Tacc * bg[1];
        out[o+2] = cb + Tacc * bg[2];
    }
}

// ---------------------------------------------------------------------------
extern "C" void kernel_launch(void* const* d_in, const int* in_sizes, int n_in,
                              void* d_out, int out_size, void* d_ws, size_t ws_size,
                              hipStream_t stream)
{
    const float* means   = (const float*)d_in[0];
    const float* scales  = (const float*)d_in[1];
    const float* quats   = (const float*)d_in[2];
    const float* fdc     = (const float*)d_in[3];
    const float* frest   = (const float*)d_in[4];
    const float* opac    = (const float*)d_in[5];
    const float* viewmat = (const float*)d_in[6];
    const float* Km      = (const float*)d_in[7];
    const float* bg      = (const float*)d_in[8];
    float* out = (float*)d_out;

    int N = in_sizes[0] / 3;
    const int H = 224, W = 224;

    // Workspace layout (floats), 16B-aligned sections.
    float* ws   = (float*)d_ws;
    size_t o0 = 0;                          // zbuf  : N
    size_t o1 = (size_t)((N + 15) & ~15);   // coef  : 8N
    size_t o2 = o1 + (size_t)8 * N;         // col   : 4N
    size_t o3 = o2 + (size_t)4 * N;         // scoef : 8N
    size_t o4 = o3 + (size_t)8 * N;         // scol  : 4N
    float* zbuf  = ws + o0;
    float* coef  = ws + o1;
    float* colb  = ws + o2;
    float* scoef = ws + o3;
    float* scol  = ws + o4;

    gs_preprocess<<<(N + TPB - 1) / TPB, TPB, 0, stream>>>(
        means, scales, quats, fdc, frest, opac, viewmat, Km,
        zbuf, coef, colb, N);

    gs_sort_gather<<<1, 1024, 0, stream>>>(zbuf, coef, colb, scoef, scol, N);

    dim3 grid((W + 15) / 16, (H + 15) / 16);
    gs_raster<<<grid, TPB, 0, stream>>>(scoef, scol, bg, out, N, W, H);
}